// Net_73220602462691
// MI455X (gfx1250) — compile-verified
//
#include <hip/hip_runtime.h>

typedef float v2f __attribute__((ext_vector_type(2)));
typedef float v8f __attribute__((ext_vector_type(8)));

// ---------------- utility kernels ----------------

__global__ void fill_kernel(float* __restrict__ p, float v, long n) {
    long i = (long)blockIdx.x * blockDim.x + threadIdx.x;
    if (i < n) p[i] = v;
}

__global__ void degree_kernel(const long long* __restrict__ dst,
                              float* __restrict__ deg, int E) {
    int e = blockIdx.x * blockDim.x + threadIdx.x;
    if (e < E) atomicAdd(&deg[(int)dst[e]], 1.0f);
}

__global__ void rsqrt_kernel(float* __restrict__ deg, int N) {
    int i = blockIdx.x * blockDim.x + threadIdx.x;
    if (i < N) {
        float d = deg[i];
        deg[i] = (d > 0.0f) ? rsqrtf(d) : 0.0f;
    }
}

// Precompute per-edge int32 endpoints + symmetric norm dis[s]*dis[d].
__global__ void edge_prep_kernel(const long long* __restrict__ src,
                                 const long long* __restrict__ dst,
                                 const float* __restrict__ dis,
                                 int* __restrict__ src32, int* __restrict__ dst32,
                                 float* __restrict__ norm, int E) {
    int e = blockIdx.x * blockDim.x + threadIdx.x;
    if (e >= E) return;
    int s = (int)src[e], d = (int)dst[e];
    src32[e] = s;
    dst32[e] = d;
    norm[e] = dis[s] * dis[d];
}

// ---------------- WMMA fp32 GEMM: XW[N,Cout] = X[N,K] @ W[K,Cout] ----------
// Compile-time K/COUT. W staged into LDS (zero-padded to KPAD rows) with
// gfx1250 async global->LDS copies; all fragment loads are unconditional so
// the fully unrolled K-loop is straight-line v_wmma_f32_16x16x4_f32 code with
// EXEC all-ones throughout (ISA requirement).
// A (16x4 f32): lanes 0-15  -> M=lane,    vgpr {K=k,  K=k+1}
//               lanes 16-31 -> M=lane-16, vgpr {K=k+2,K=k+3}
// B (4x16 f32): lanes 0-15 N=lane rows K=k/K=k+1; lanes 16-31 rows K=k+2/K=k+3
// C/D (16x16 f32): vgpr r -> M = r + 8*half, N = lane&15.

template <int K, int COUT>
__global__ void gemm_wmma_kernel(const float* __restrict__ X,
                                 const float* __restrict__ Wg,
                                 float* __restrict__ XW, int N) {
    constexpr int KPAD = (K + 3) & ~3;
    constexpr int CT   = COUT / 16;
    __shared__ float Wlds[KPAD * COUT];   // only LDS object -> LDS offset 0

    // --- async stage W (K*COUT floats) into LDS; zero the K..KPAD pad rows ---
    for (int i = threadIdx.x * 4; i < K * COUT; i += blockDim.x * 4) {
        unsigned ofs = (unsigned)(i * 4);            // bytes; LDS dest == global src offset
        asm volatile("global_load_async_to_lds_b128 %0, %1, %2"
                     :: "v"(ofs), "v"(ofs), "s"(Wg)
                     : "memory");
    }
    for (int i = K * COUT + threadIdx.x; i < KPAD * COUT; i += blockDim.x)
        Wlds[i] = 0.0f;
    asm volatile("s_wait_asynccnt 0x0" ::: "memory");
    __syncthreads();

    const int wave  = (int)((blockIdx.x * blockDim.x + threadIdx.x) >> 5);
    const int lane  = threadIdx.x & 31;
    const int tiles = (N + 15) >> 4;
    if (wave >= tiles) return;            // uniform per-wave exit (after barrier)

    const int m0   = wave << 4;
    const int half = lane >> 4;           // 0 or 1
    const int mrow = lane & 15;

    v8f acc[CT];
    const v8f vzero = {0.f, 0.f, 0.f, 0.f, 0.f, 0.f, 0.f, 0.f};
#pragma unroll
    for (int t = 0; t < CT; ++t) acc[t] = vzero;

    int arow = m0 + mrow;
    if (arow > N - 1) arow = N - 1;       // clamp (pad rows never stored)
    const float* Xrow = X + (long)arow * K;

#pragma unroll
    for (int k = 0; k < KPAD; k += 4) {
        const int ka = k + 2 * half;
        // clamped in-bounds loads; padded region multiplies zero B -> no effect
        const int ia = (ka     < K) ? ka     : (K - 1);
        const int ib = (ka + 1 < K) ? ka + 1 : (K - 1);
        v2f a;
        a.x = Xrow[ia];
        a.y = Xrow[ib];
#pragma unroll
        for (int t = 0; t < CT; ++t) {
            const int col = t * 16 + mrow;
            v2f b;
            b.x = Wlds[ka * COUT + col];
            b.y = Wlds[(ka + 1) * COUT + col];
            acc[t] = __builtin_amdgcn_wmma_f32_16x16x4_f32(
                false, a, false, b, (short)0, acc[t], false, false);
        }
    }

#pragma unroll
    for (int t = 0; t < CT; ++t) {
        const int col = t * 16 + mrow;
#pragma unroll
        for (int r = 0; r < 8; ++r) {
            const int row = m0 + r + 8 * half;
            if (row < N) XW[(long)row * COUT + col] = acc[t][r];
        }
    }
}

// ---------------- edge scatter: agg[dst] += norm[e] * xw[src] --------------
// C/4 threads per edge, float4 gather, 4x global_atomic_add_f32 scatter.

template <int C>
__global__ void scatter_kernel(const int* __restrict__ src32,
                               const int* __restrict__ dst32,
                               const float* __restrict__ norm,
                               const float* __restrict__ xw,
                               float* __restrict__ agg, int E) {
    const int TPE = C / 4;
    long gid = (long)blockIdx.x * blockDim.x + threadIdx.x;
    int e  = (int)(gid / TPE);
    int fc = (int)(gid % TPE) * 4;
    if (e >= E) return;
    const int   s   = src32[e];
    const int   d   = dst32[e];
    const float nrm = norm[e];
    const float4 v = *reinterpret_cast<const float4*>(xw + (long)s * C + fc);
    float* o = agg + (long)d * C + fc;
    atomicAdd(o + 0, nrm * v.x);
    atomicAdd(o + 1, nrm * v.y);
    atomicAdd(o + 2, nrm * v.z);
    atomicAdd(o + 3, nrm * v.w);
}

// --------------- epilogue: self-loop + bias + relu, in place on agg --------

__global__ void bias_relu_self_kernel(float* __restrict__ agg,
                                      const float* __restrict__ xw,
                                      const float* __restrict__ dis,
                                      const float* __restrict__ b,
                                      int N, int C) {
    long i = (long)blockIdx.x * blockDim.x + threadIdx.x;
    if (i >= (long)N * C) return;
    const int node = (int)(i / C);
    const int c    = (int)(i % C);
    const float ds = dis[node];
    float v = agg[i] + ds * ds * xw[i] + b[c];
    agg[i] = (v > 0.0f) ? v : 0.0f;
}

// ----- fused layer-2 epilogue + final linear:
//       out[i] = relu(agg2[i,:] + dis^2*xw2[i,:] + b2) . Wl + bl -------------

__global__ void fused_final_kernel(const float* __restrict__ agg2,
                                   const float* __restrict__ xw2,
                                   const float* __restrict__ dis,
                                   const float* __restrict__ b2,
                                   const float* __restrict__ Wl,
                                   const float* __restrict__ bl,
                                   float* __restrict__ out, int N) {
    int i = blockIdx.x * blockDim.x + threadIdx.x;
    if (i >= N) return;
    const float ds  = dis[i];
    const float ds2 = ds * ds;
    float s = bl[0];
    const float* ar = agg2 + (long)i * 32;
    const float* xr = xw2 + (long)i * 32;
#pragma unroll
    for (int c = 0; c < 32; c += 4) {
        float4 a  = *reinterpret_cast<const float4*>(ar + c);
        float4 xv = *reinterpret_cast<const float4*>(xr + c);
        float4 bv = *reinterpret_cast<const float4*>(b2 + c);
        float4 wv = *reinterpret_cast<const float4*>(Wl + c);
        s += fmaxf(a.x + ds2 * xv.x + bv.x, 0.0f) * wv.x;
        s += fmaxf(a.y + ds2 * xv.y + bv.y, 0.0f) * wv.y;
        s += fmaxf(a.z + ds2 * xv.z + bv.z, 0.0f) * wv.z;
        s += fmaxf(a.w + ds2 * xv.w + bv.w, 0.0f) * wv.w;
    }
    out[i] = s;
}

// ---------------------------------------------------------------------------

static inline int cdiv(long a, int b) { return (int)((a + b - 1) / b); }

extern "C" void kernel_launch(void* const* d_in, const int* in_sizes, int n_in,
                              void* d_out, int out_size, void* d_ws, size_t ws_size,
                              hipStream_t stream) {
    const float*     x   = (const float*)d_in[0];
    const long long* ei  = (const long long*)d_in[1];
    const float*     W1  = (const float*)d_in[2];
    const float*     b1  = (const float*)d_in[3];
    const float*     W2  = (const float*)d_in[4];
    const float*     b2  = (const float*)d_in[5];
    const float*     Wl  = (const float*)d_in[6];
    const float*     bl  = (const float*)d_in[7];
    float*           out = (float*)d_out;

    const int N = in_sizes[0] / 26;
    const int E = in_sizes[1] / 2;
    const long long* src = ei;
    const long long* dst = ei + E;

    // workspace layout (floats):
    //   dis[N] | bufA[N*64] (agg1 -> h1 in place) |
    //   bufB[N*64] (xw1; then xw2 in lo half, agg2 in hi half) |
    //   src32[E] | dst32[E] | norm[E]
    float* ws    = (float*)d_ws;
    float* dis   = ws;
    float* bufA  = ws + (((size_t)N + 63) & ~(size_t)63);
    float* bufB  = bufA + (size_t)N * 64;
    int*   src32 = (int*)(bufB + (size_t)N * 64);
    int*   dst32 = src32 + E;
    float* norm  = (float*)(dst32 + E);

    float* xw1  = bufB;
    float* agg1 = bufA;                      // becomes h1 in place
    float* xw2  = bufB;                      // first N*32 floats
    float* agg2 = bufB + (size_t)N * 32;

    const int BLK = 256;

    // 1) deg = 1.0 (self loop), agg1 = 0
    fill_kernel<<<cdiv(N, BLK), BLK, 0, stream>>>(dis, 1.0f, (long)N);
    fill_kernel<<<cdiv((long)N * 64, BLK), BLK, 0, stream>>>(agg1, 0.0f, (long)N * 64);

    // 2) degrees on A + I, dis = rsqrt(deg), per-edge norms + int32 endpoints
    degree_kernel<<<cdiv(E, BLK), BLK, 0, stream>>>(dst, dis, E);
    rsqrt_kernel<<<cdiv(N, BLK), BLK, 0, stream>>>(dis, N);
    edge_prep_kernel<<<cdiv(E, BLK), BLK, 0, stream>>>(src, dst, dis, src32, dst32, norm, E);

    // 3) xw1 = x @ W1   [N,26]@[26,64]  (WMMA fp32, 4 col tiles)
    {
        const int waves  = (N + 15) / 16;
        const int blocks = cdiv((long)waves * 32, BLK);
        gemm_wmma_kernel<26, 64><<<blocks, BLK, 0, stream>>>(x, W1, xw1, N);
    }

    // 4) scatter layer 1 (16 threads/edge, 4 feats each)
    scatter_kernel<64><<<cdiv((long)E * 16, BLK), BLK, 0, stream>>>(src32, dst32, norm, xw1, agg1, E);

    // 5) h1 = relu(agg1 + dis^2 * xw1 + b1)  (in place on agg1)
    bias_relu_self_kernel<<<cdiv((long)N * 64, BLK), BLK, 0, stream>>>(agg1, xw1, dis, b1, N, 64);

    // 6) agg2 = 0 (xw1 dead; hi half of bufB reused)
    fill_kernel<<<cdiv((long)N * 32, BLK), BLK, 0, stream>>>(agg2, 0.0f, (long)N * 32);

    // 7) xw2 = h1 @ W2   [N,64]@[64,32]  (WMMA fp32, 2 col tiles)
    {
        const int waves  = (N + 15) / 16;
        const int blocks = cdiv((long)waves * 32, BLK);
        gemm_wmma_kernel<64, 32><<<blocks, BLK, 0, stream>>>(agg1, W2, xw2, N);
    }

    // 8) scatter layer 2 (8 threads/edge)
    scatter_kernel<32><<<cdiv((long)E * 8, BLK), BLK, 0, stream>>>(src32, dst32, norm, xw2, agg2, E);

    // 9) out = relu(agg2 + dis^2*xw2 + b2) @ Wl + bl   (fused, no h2 round-trip)
    fused_final_kernel<<<cdiv(N, BLK), BLK, 0, stream>>>(agg2, xw2, dis, b2, Wl, bl, out, N);
}